// Encoder_19662360281723
// MI455X (gfx1250) — compile-verified
//
#include <hip/hip_runtime.h>
#include <hip/hip_bf16.h>

typedef __attribute__((ext_vector_type(2))) float v2f;
typedef __attribute__((ext_vector_type(8))) float v8f;

#define BN_EPS 1e-5f

// ---------------------------------------------------------------------------
// 1) Embedding gather + sum-pool.  One block = one (row, side). 128 threads
//    cover 300 columns in 3 strided chunks -> each token row (1200 B) is read
//    fully coalesced. tables[:,0,:] is already zero (padding_idx).
// ---------------------------------------------------------------------------
__global__ void k_gather(const int* __restrict__ langs1, const int* __restrict__ sents1,
                         const int* __restrict__ langs2, const int* __restrict__ sents2,
                         const float* __restrict__ tables, float* __restrict__ X,
                         int B, int S, int V, int D) {
  const int m = blockIdx.x;
  const int side = blockIdx.y;
  const int t = threadIdx.x;  // 128 threads
  const int* langs = side ? langs2 : langs1;
  const int* sents = side ? sents2 : sents1;
  const size_t lang = (size_t)langs[m];
  const int* toks = sents + (size_t)m * S;
  float a0 = 0.f, a1 = 0.f, a2 = 0.f;
  for (int s = 0; s < S; ++s) {
    const size_t tok = (size_t)toks[s];
    const float* row = tables + (lang * (size_t)V + tok) * (size_t)D;
    a0 += row[t];
    a1 += row[t + 128];
    if (t < D - 256) a2 += row[t + 256];
  }
  float* out = X + ((size_t)side * B + m) * (size_t)D;
  out[t] = a0;
  out[t + 128] = a1;
  if (t < D - 256) out[t + 256] = a2;
}

// ---------------------------------------------------------------------------
// 2) Generic GEMM  Y[M][N] = A[M][K] * W[N][K]^T + bias[N]   via
//    V_WMMA_F32_16X16X4_F32.  Block = 256 threads = 8 waves; each wave owns a
//    16x16 tile (wave w -> tile row blockIdx.y*8+w, tile col blockIdx.x).
//    Operand layouts (wave32):
//      A 16x4 : lane l+16*half holds A[row0+l][k+2*half + {0,1}] in v[0],v[1]
//      B 4x16 : lane l+16*half holds W[col0+l][k+2*half + {0,1}] (B = W^T)
//      C 16x16: v[r] on lane l+16*half = D[row0 + r + 8*half][col0 + l]
//    EXEC stays all-ones: the only pre-WMMA exits are wave-uniform, and the
//    N-edge is handled by clamping the W row (results masked at the store).
// ---------------------------------------------------------------------------
__global__ void k_gemm_wmma(const float* __restrict__ A, int lda,
                            const float* __restrict__ W, int ldw,
                            const float* __restrict__ bias,
                            float* __restrict__ Y, int ldy,
                            int M, int N, int K) {
  const int lane = threadIdx.x & 31;
  const int wave = threadIdx.x >> 5;
  const int half = lane >> 4;
  const int l = lane & 15;
  const int row0 = (blockIdx.y * 8 + wave) * 16;
  const int col0 = blockIdx.x * 16;
  if (row0 >= M) return;                       // wave-uniform
  const int ar = row0 + l;                     // M is a multiple of 16
  const int wr = min(col0 + l, N - 1);         // clamp keeps all lanes active
  const float* Ap = A + (size_t)ar * lda;
  const float* Wp = W + (size_t)wr * ldw;

  v8f c = {};
  const int K4 = K & ~3;
  int k = 0;
  for (; k < K4; k += 4) {
    const int kk = k + 2 * half;
    v2f a, b;
    a.x = Ap[kk]; a.y = Ap[kk + 1];            // b32 loads: safe for odd ldw
    b.x = Wp[kk]; b.y = Wp[kk + 1];
    c = __builtin_amdgcn_wmma_f32_16x16x4_f32(false, a, false, b,
                                              (short)0, c, false, false);
  }
  if (k < K) {                                 // K remainder (K=901 case)
    const int kk = k + 2 * half;
    v2f a, b;
    a.x = (kk     < K) ? Ap[kk]     : 0.f;
    a.y = (kk + 1 < K) ? Ap[kk + 1] : 0.f;
    b.x = (kk     < K) ? Wp[kk]     : 0.f;
    b.y = (kk + 1 < K) ? Wp[kk + 1] : 0.f;
    c = __builtin_amdgcn_wmma_f32_16x16x4_f32(false, a, false, b,
                                              (short)0, c, false, false);
  }

  const int col = col0 + l;
  if (col < N) {
    const float bv = bias ? bias[col] : 0.f;
#pragma unroll
    for (int r = 0; r < 8; ++r) {
      const int mrow = row0 + r + 8 * half;
      Y[(size_t)mrow * ldy + col] = c[r] + bv;
    }
  }
}

// ---------------------------------------------------------------------------
// 3) Per-column batch statistics (biased variance, matching jnp.var).
//    One thread per column; row loop is coalesced across the block.
// ---------------------------------------------------------------------------
__global__ void k_stats(const float* __restrict__ Y, int ld,
                        float* __restrict__ mean, float* __restrict__ rstd,
                        int rows, int cols) {
  const int c = blockIdx.x * blockDim.x + threadIdx.x;
  const int seg = blockIdx.y;
  if (c >= cols) return;
  const float* p = Y + (size_t)seg * rows * ld + c;
  float s = 0.f, s2 = 0.f;
  for (int r = 0; r < rows; ++r) {
    const float v = p[(size_t)r * ld];
    s += v; s2 += v * v;
  }
  const float mu = s / (float)rows;
  const float var = s2 / (float)rows - mu * mu;
  mean[seg * cols + c] = mu;
  rstd[seg * cols + c] = rsqrtf(var + BN_EPS);
}

// ---------------------------------------------------------------------------
// 4) BatchNorm(affine=False) + ReLU, elementwise.
// ---------------------------------------------------------------------------
__global__ void k_bnrelu(const float* __restrict__ Y, int ld,
                         const float* __restrict__ mean, const float* __restrict__ rstd,
                         float* __restrict__ O, int ldo,
                         int rows, int cols) {
  const int i = blockIdx.x * blockDim.x + threadIdx.x;
  const int seg = blockIdx.y;
  if (i >= rows * cols) return;
  const int r = i / cols, c = i % cols;
  const float v = Y[(size_t)seg * rows * ld + (size_t)r * ld + c];
  const float z = (v - mean[seg * cols + c]) * rstd[seg * cols + c];
  O[(size_t)seg * rows * ldo + (size_t)r * ldo + c] = fmaxf(z, 0.f);
}

// ---------------------------------------------------------------------------
// 5) Feature builder: nxt[m] = [o1, o2, |o1-o2|, dot(o1,o2)], stride-padded.
// ---------------------------------------------------------------------------
__global__ void k_feat(const float* __restrict__ O, float* __restrict__ nxt,
                       int B, int D, int ldn) {
  const int m = blockIdx.x;
  const int t = threadIdx.x;  // 256
  const float* o1 = O + (size_t)m * D;
  const float* o2 = O + ((size_t)B + m) * D;
  float* out = nxt + (size_t)m * ldn;
  __shared__ float red[256];
  float part = 0.f;
  for (int c = t; c < D; c += 256) {
    const float a = o1[c], b = o2[c];
    out[c] = a;
    out[D + c] = b;
    out[2 * D + c] = fabsf(a - b);
    part += a * b;
  }
  red[t] = part;
  __syncthreads();
  for (int off = 128; off > 0; off >>= 1) {
    if (t < off) red[t] += red[t + off];
    __syncthreads();
  }
  if (t == 0) {
    out[3 * D] = red[0];
    out[3 * D + 1] = 0.f; out[3 * D + 2] = 0.f; out[3 * D + 3] = 0.f;
  }
}

// ---------------------------------------------------------------------------
// 6) Final dot (w5: [1,100]) + sigmoid.  One wave32 per row, shfl reduction.
// ---------------------------------------------------------------------------
__global__ void k_final(const float* __restrict__ H, const float* __restrict__ w5,
                        const float* __restrict__ b5, float* __restrict__ prob,
                        int B, int C) {
  const int lane = threadIdx.x & 31;
  const int wave = threadIdx.x >> 5;
  const int m = blockIdx.x * 4 + wave;
  if (m >= B) return;
  const float* h = H + (size_t)m * C;
  float part = 0.f;
  for (int c = lane; c < C; c += 32) part += h[c] * w5[c];
  for (int off = 16; off > 0; off >>= 1) part += __shfl_xor(part, off, 32);
  if (lane == 0) {
    const float s = part + b5[0];
    prob[m] = 1.f / (1.f + expf(-s));
  }
}

// ---------------------------------------------------------------------------
extern "C" void kernel_launch(void* const* d_in, const int* in_sizes, int n_in,
                              void* d_out, int out_size, void* d_ws, size_t ws_size,
                              hipStream_t stream) {
  const int B = 1024, S = 50, V = 200000, D = 300;
  const int* langs1 = (const int*)d_in[0];
  const int* sents1 = (const int*)d_in[1];
  const int* langs2 = (const int*)d_in[2];
  const int* sents2 = (const int*)d_in[3];
  const float* tables = (const float*)d_in[4];
  const float* w1 = (const float*)d_in[5];
  const float* b1 = (const float*)d_in[6];
  const float* w2 = (const float*)d_in[7];
  const float* b2 = (const float*)d_in[8];
  const float* w3 = (const float*)d_in[9];
  const float* b3 = (const float*)d_in[10];
  const float* w4 = (const float*)d_in[11];
  const float* b4 = (const float*)d_in[12];
  const float* w5 = (const float*)d_in[13];
  const float* b5 = (const float*)d_in[14];

  float* ws = (float*)d_ws;
  // Workspace layout (floats); stages reuse regions only after consumers ran.
  float* X    = ws;                 // [2][1024][300] gather out, then H1
  float* Ybuf = ws + 614400;        // [2][1024][300] pre-BN activations
  float* nxt  = ws;                 // [1024][904]  (X/H1 dead by then)
  float* Z3   = ws + 1000000;       // [1024][400]
  float* H3   = ws + 1500000;       // [1024][400]
  float* Z4   = ws + 2000000;       // [1024][100]
  float* H4   = ws + 2200000;       // [1024][100]
  float* mean = ws + 2400000;       // up to 2*400
  float* rstd = ws + 2450000;       // up to 2*400

  float* prob = (float*)d_out;            // [1024]
  float* O    = (float*)d_out + 1024;     // [2][1024][300] = o1 then o2

  // 1) gather + sum-pool
  k_gather<<<dim3(B, 2), 128, 0, stream>>>(langs1, sents1, langs2, sents2,
                                           tables, X, B, S, V, D);

  // 2) layer1: Y = X @ w1^T + b1  (M=2048 rows = both sides stacked)
  k_gemm_wmma<<<dim3(19, 16), 256, 0, stream>>>(X, 300, w1, 300, b1,
                                                Ybuf, 300, 2048, 300, 300);
  k_stats<<<dim3(2, 2), 256, 0, stream>>>(Ybuf, 300, mean, rstd, 1024, 300);
  k_bnrelu<<<dim3(1200, 2), 256, 0, stream>>>(Ybuf, 300, mean, rstd,
                                              X, 300, 1024, 300);  // H1 -> X

  // 3) layer2: Y = H1 @ w2^T + b2 ; BN+ReLU straight into d_out (o1,o2)
  k_gemm_wmma<<<dim3(19, 16), 256, 0, stream>>>(X, 300, w2, 300, b2,
                                                Ybuf, 300, 2048, 300, 300);
  k_stats<<<dim3(2, 2), 256, 0, stream>>>(Ybuf, 300, mean, rstd, 1024, 300);
  k_bnrelu<<<dim3(1200, 2), 256, 0, stream>>>(Ybuf, 300, mean, rstd,
                                              O, 300, 1024, 300);

  // 4) features: [o1, o2, |o1-o2|, cor] -> nxt[1024][904] (K=901 used)
  k_feat<<<dim3(B), 256, 0, stream>>>(O, nxt, B, 300, 904);

  // 5) layer3: Z3 = nxt @ w3^T + b3  (K=901, odd ldw handled by b32 loads)
  k_gemm_wmma<<<dim3(25, 8), 256, 0, stream>>>(nxt, 904, w3, 901, b3,
                                               Z3, 400, 1024, 400, 901);
  k_stats<<<dim3(2, 1), 256, 0, stream>>>(Z3, 400, mean, rstd, 1024, 400);
  k_bnrelu<<<dim3(1600, 1), 256, 0, stream>>>(Z3, 400, mean, rstd,
                                              H3, 400, 1024, 400);

  // 6) layer4: Z4 = H3 @ w4^T + b4
  k_gemm_wmma<<<dim3(7, 8), 256, 0, stream>>>(H3, 400, w4, 400, b4,
                                              Z4, 100, 1024, 100, 400);
  k_stats<<<dim3(1, 1), 256, 0, stream>>>(Z4, 100, mean, rstd, 1024, 100);
  k_bnrelu<<<dim3(400, 1), 256, 0, stream>>>(Z4, 100, mean, rstd,
                                             H4, 100, 1024, 100);

  // 7) score + sigmoid
  k_final<<<dim3(256), 128, 0, stream>>>(H4, w5, b5, prob, B, 100);
}